// CGLayer_18227841204318
// MI455X (gfx1250) — compile-verified
//
#include <hip/hip_runtime.h>
#include <stdint.h>

typedef __attribute__((ext_vector_type(2))) float v2f;
typedef __attribute__((ext_vector_type(8))) float v8f;

#define NL 6          // L_MAX+1
#define CIN 24
#define HID 32
#define BT 16         // batch tile
#define DTOT 36       // sum of (2l+1)

// pair tables: for each l, (l1,l2) with |l1-l2| <= l <= l1+l2, l1-major order
__device__ __constant__ int PAIR_START_C[6] = {0, 6, 21, 42, 66, 90};
__device__ __constant__ int PAIR_COUNT_C[6] = {6, 15, 21, 24, 24, 21};
__device__ __constant__ signed char PAIR_L1_C[111] = {
    // l=0
    0,1,2,3,4,5,
    // l=1
    0,1,1,1,2,2,2,3,3,3,4,4,4,5,5,
    // l=2
    0,1,1,1,2,2,2,2,2,3,3,3,3,3,4,4,4,4,5,5,5,
    // l=3
    0,1,1,1,2,2,2,2,2,3,3,3,3,3,3,4,4,4,4,4,5,5,5,5,
    // l=4
    0,1,1,1,2,2,2,2,3,3,3,3,3,4,4,4,4,4,4,5,5,5,5,5,
    // l=5
    0,1,1,2,2,2,3,3,3,3,4,4,4,4,4,5,5,5,5,5,5
};
__device__ __constant__ signed char PAIR_L2_C[111] = {
    // l=0
    0,1,2,3,4,5,
    // l=1
    1,0,1,2,1,2,3,2,3,4,3,4,5,4,5,
    // l=2
    2,1,2,3,0,1,2,3,4,1,2,3,4,5,2,3,4,5,3,4,5,
    // l=3
    3,2,3,4,1,2,3,4,5,0,1,2,3,4,5,1,2,3,4,5,2,3,4,5,
    // l=4
    4,3,4,5,2,3,4,5,1,2,3,4,5,0,1,2,3,4,5,1,2,3,4,5,
    // l=5
    5,4,5,3,4,5,2,3,4,5,1,2,3,4,5,0,1,2,3,4,5
};
// LDS x layout: per l, [b][c*(2l+1)+i] with per-b stride 24*(2l+1)+1 (bank-conflict pad)
__device__ __constant__ int XOFF_C[6] = {0, 400, 1568, 3504, 6208, 9680};  // floats
__device__ __constant__ int XSTR_C[6] = {25, 73, 121, 169, 217, 265};
__device__ __constant__ int MOFF_C[6] = {0, 1, 4, 9, 16, 25};

static __device__ __forceinline__ int imaxi(int a, int b) { return a > b ? a : b; }
static __device__ __forceinline__ int imini(int a, int b) { return a < b ? a : b; }

__global__ __launch_bounds__(256) void cg_tp_wmma_kernel(
    const float* __restrict__ x0, const float* __restrict__ x1,
    const float* __restrict__ x2, const float* __restrict__ x3,
    const float* __restrict__ x4, const float* __restrict__ x5,
    const float* __restrict__ W0, const float* __restrict__ W1,
    const float* __restrict__ W2, const float* __restrict__ W3,
    const float* __restrict__ W4, const float* __restrict__ W5,
    const float* __restrict__ cg, float* __restrict__ out)
{
    __shared__ float xs_lds[13920];   // all x_l for this batch tile (padded)
    __shared__ float cg_lds[1331];    // 11x11x11 cg slice for current (l1,l2,l)
    __shared__ float w_lds[32 * 32];  // W chunk: 32 h x 32 k

    const int tid   = threadIdx.x;
    const int lane  = tid & 31;
    const int wv    = tid >> 5;       // wave id 0..7
    const int khalf = lane >> 4;      // 0: lanes 0-15, 1: lanes 16-31
    const int nl    = lane & 15;      // M for A-frag; N (batch) for B/C/D frags

    const int bt   = blockIdx.x;      // batch tile 0..15
    const int l    = blockIdx.y;      // output order 0..5
    const int twol = 2 * l;

    // ---- stage all x_l tiles (16 batches) into LDS: straight-line rounds + tail ----
    {
        const float* xsrc[6] = {x0, x1, x2, x3, x4, x5};
#pragma unroll
        for (int lx = 0; lx < 6; ++lx) {
            const int dim    = 2 * lx + 1;
            const int row    = CIN * dim;        // contiguous floats per batch
            const int cnt    = BT * row;         // == rounds*256 + 128 for all lx
            const int rounds = cnt >> 8;
            const int str    = row + 1;
            const float* src = xsrc[lx] + (size_t)bt * BT * row;
            const int off    = XOFF_C[lx];
#pragma unroll
            for (int r = 0; r < rounds; ++r) {
                int idx = tid + r * 256;
                int b   = idx / row;
                int rem = idx - b * row;
                xs_lds[off + b * str + rem] = src[idx];
            }
            if (tid < 128) {
                int idx = tid + rounds * 256;
                int b   = idx / row;
                int rem = idx - b * row;
                xs_lds[off + b * str + rem] = src[idx];
            }
        }
    }

    const float* Wl = (l == 0) ? W0 : (l == 1) ? W1 : (l == 2) ? W2
                    : (l == 3) ? W3 : (l == 4) ? W4 : W5;
    const int npair  = PAIR_COUNT_C[l];
    const int pstart = PAIR_START_C[l];
    const int k_in   = npair * (CIN * CIN);

    // LDS byte offset of w_lds (flat LDS aperture keeps offset in addr[31:0])
    const uint32_t w_lds_base = (uint32_t)(uintptr_t)(void*)w_lds;
    // per-thread slot for async W staging: thread t moves dwords [4t, 4t+3]
    const uint32_t w_lds_dst  = w_lds_base + ((uint32_t)tid << 4);
    const int      wh         = tid >> 3;          // h row 0..31
    const int      wkk        = (tid & 7) << 2;    // k col 0,4,...,28
    const uint32_t w_voff     = (uint32_t)((wh * k_in + wkk) * 4);  // global byte offset

    // wave -> m assignment (<= 2 m's per wave)
    const int  m0   = wv;
    const int  m1   = wv + 8;
    const bool has0 = (m0 <= twol);
    const bool has1 = (m1 <= twol);

    v8f acc00 = {0.f,0.f,0.f,0.f,0.f,0.f,0.f,0.f};  // m0, h 0..15
    v8f acc01 = acc00;                              // m0, h 16..31
    v8f acc10 = acc00;                              // m1, h 0..15
    v8f acc11 = acc00;                              // m1, h 16..31

    for (int pp = 0; pp < npair; ++pp) {
        const int pidx  = pstart + pp;
        const int l1    = PAIR_L1_C[pidx];
        const int l2    = PAIR_L2_C[pidx];
        const int d1dim = 2 * l1 + 1;
        const int d2dim = 2 * l2 + 1;
        const int x1off = XOFF_C[l1], x1str = XSTR_C[l1];
        const int x2off = XOFF_C[l2], x2str = XSTR_C[l2];
        const float* cgsl = cg + (size_t)(((l1 * 6 + l2) * 6 + l)) * 1331;

        __syncthreads();  // previous pair fully consumed cg_lds / w_lds (and x load visible)
        // cg slice: 1331 = 5*256 + 51, straight-line
#pragma unroll
        for (int r = 0; r < 5; ++r) cg_lds[tid + r * 256] = cgsl[tid + r * 256];
        if (tid < 51) cg_lds[1280 + tid] = cgsl[1280 + tid];

        // diagonal bounds: i + j = S,  i in [max(0,S-2l2), min(2l1,S)]
        const int S0   = m0 + l1 + l2 - l;
        const int S1   = m1 + l1 + l2 - l;
        const int i0lo = imaxi(0, S0 - 2 * l2), i0hi = imini(2 * l1, S0);
        const int i1lo = imaxi(0, S1 - 2 * l2), i1hi = imini(2 * l1, S1);

        const int kbase = pp * (CIN * CIN);   // K offset of this pair inside W_l
        for (int kc0 = 0; kc0 < CIN * CIN; kc0 += 32) {
            __syncthreads();  // cg_lds visible (1st iter) / prev W chunk consumed

            // ---- async global->LDS staging of W chunk (32h x 32k = 4KB) ----
            // one GLOBAL_LOAD_ASYNC_TO_LDS_B128 per thread, tracked by ASYNCcnt
            {
                const float* wsrc = Wl + kbase + kc0;
                asm volatile("global_load_async_to_lds_b128 %0, %1, %2"
                             :: "v"(w_lds_dst), "v"(w_voff), "s"(wsrc)
                             : "memory");
                asm volatile("s_wait_asynccnt 0x0" ::: "memory");
            }
            __syncthreads();

#pragma unroll
            for (int ks = 0; ks < 8; ++ks) {
                const int kk0 = ks * 4 + 2 * khalf;  // this lane's K rows: kk0, kk0+1
                // A fragments (W tile), ISA layout: lane M = lane&15, K = 2*(lane>>4)+{0,1}
                v2f aLo, aHi;
                aLo.x = w_lds[nl * 32 + kk0];
                aLo.y = w_lds[nl * 32 + kk0 + 1];
                aHi.x = w_lds[(nl + 16) * 32 + kk0];
                aHi.y = w_lds[(nl + 16) * 32 + kk0 + 1];

                // (c,d) for this lane's two K rows
                const int kloc  = kc0 + kk0;
                const int c0    = kloc / CIN, d0 = kloc - c0 * CIN;
                const int klocB = kloc + 1;
                const int c1    = klocB / CIN, d1v = klocB - c1 * CIN;

                const float* xr1a = &xs_lds[x1off + nl * x1str + c0 * d1dim];
                const float* xr1b = &xs_lds[x1off + nl * x1str + c1 * d1dim];
                const float* xr2a = &xs_lds[x2off + nl * x2str + d0 * d2dim];
                const float* xr2b = &xs_lds[x2off + nl * x2str + d1v * d2dim];

                if (has0) {  // wave-uniform -> EXEC all ones inside
                    float bx = 0.f, by = 0.f;
                    for (int i = i0lo; i <= i0hi; ++i) {
                        int   j   = S0 - i;
                        float cgv = cg_lds[(m0 * 11 + i) * 11 + j];
                        bx = fmaf(cgv * xr1a[i], xr2a[j], bx);
                        by = fmaf(cgv * xr1b[i], xr2b[j], by);
                    }
                    v2f bf; bf.x = bx; bf.y = by;
                    acc00 = __builtin_amdgcn_wmma_f32_16x16x4_f32(
                        false, aLo, false, bf, (short)0, acc00, false, false);
                    acc01 = __builtin_amdgcn_wmma_f32_16x16x4_f32(
                        false, aHi, false, bf, (short)0, acc01, false, false);
                }
                if (has1) {
                    float bx = 0.f, by = 0.f;
                    for (int i = i1lo; i <= i1hi; ++i) {
                        int   j   = S1 - i;
                        float cgv = cg_lds[(m1 * 11 + i) * 11 + j];
                        bx = fmaf(cgv * xr1a[i], xr2a[j], bx);
                        by = fmaf(cgv * xr1b[i], xr2b[j], by);
                    }
                    v2f bf; bf.x = bx; bf.y = by;
                    acc10 = __builtin_amdgcn_wmma_f32_16x16x4_f32(
                        false, aLo, false, bf, (short)0, acc10, false, false);
                    acc11 = __builtin_amdgcn_wmma_f32_16x16x4_f32(
                        false, aHi, false, bf, (short)0, acc11, false, false);
                }
            }
        }
    }

    // ---- epilogue: C/D layout -> out[b][h][moff+m], out is (256, 32, 36) ----
    const int moff  = MOFF_C[l];
    const int bglob = bt * BT + nl;
    float* orow = out + (size_t)bglob * HID * DTOT;
    if (has0) {
#pragma unroll
        for (int r = 0; r < 8; ++r) {
            int h = r + 8 * khalf;
            orow[h * DTOT + moff + m0]        = acc00[r];
            orow[(h + 16) * DTOT + moff + m0] = acc01[r];
        }
    }
    if (has1) {
#pragma unroll
        for (int r = 0; r < 8; ++r) {
            int h = r + 8 * khalf;
            orow[h * DTOT + moff + m1]        = acc10[r];
            orow[(h + 16) * DTOT + moff + m1] = acc11[r];
        }
    }
}

extern "C" void kernel_launch(void* const* d_in, const int* in_sizes, int n_in,
                              void* d_out, int out_size, void* d_ws, size_t ws_size,
                              hipStream_t stream) {
    (void)in_sizes; (void)n_in; (void)d_ws; (void)ws_size; (void)out_size;
    const float* x0 = (const float*)d_in[0];
    const float* x1 = (const float*)d_in[1];
    const float* x2 = (const float*)d_in[2];
    const float* x3 = (const float*)d_in[3];
    const float* x4 = (const float*)d_in[4];
    const float* x5 = (const float*)d_in[5];
    const float* W0 = (const float*)d_in[6];
    const float* W1 = (const float*)d_in[7];
    const float* W2 = (const float*)d_in[8];
    const float* W3 = (const float*)d_in[9];
    const float* W4 = (const float*)d_in[10];
    const float* W5 = (const float*)d_in[11];
    const float* cg = (const float*)d_in[12];
    float* out = (float*)d_out;

    dim3 grid(256 / BT, NL, 1);  // 16 batch tiles x 6 l-values
    dim3 block(256, 1, 1);       // 8 waves
    cg_tp_wmma_kernel<<<grid, block, 0, stream>>>(
        x0, x1, x2, x3, x4, x5, W0, W1, W2, W3, W4, W5, cg, out);
}